// labelwisepassing_61770219651594
// MI455X (gfx1250) — compile-verified
//
#include <hip/hip_runtime.h>

#define NN 16384
#define DD 512
#define HH 64
#define KCH 8          // split-K chunks for the aggregation GEMM
#define KLEN (NN / KCH)

typedef __attribute__((ext_vector_type(2))) float v2f;
typedef __attribute__((ext_vector_type(4))) float v4f;
typedef __attribute__((ext_vector_type(8))) float v8f;

// Workspace layout (float offsets):
//   [0, 16384)          deg[N]
//   [16384, 16400)      s[16]            (label sums, only 0..6 used)
//   [16400, 24592)      Amat[16][512]
//   [24592, 90128)      partial[8][16][512]   (split-K partial tiles)
//   [90128, 352272)     coefT[16][N]     (only rows 0..7 written; rows 8..15
//                                         feed discarded WMMA output rows)
// Total ~1.41 MB.

__global__ void zero_s_kernel(float* __restrict__ s) {
  if (threadIdx.x < 16) s[threadIdx.x] = 0.f;
}

// deg[row] = sum of matrix row; 1.07 GB streamed once with NT hint (don't
// evict xf/coef from the 192 MB L2 — the matrix is touched exactly once).
__global__ __launch_bounds__(256) void deg_kernel(const float* __restrict__ matrix,
                                                  float* __restrict__ deg) {
  const int row = blockIdx.x;
  const v4f* m4 = reinterpret_cast<const v4f*>(matrix + (size_t)row * NN);
  float acc = 0.f;
  for (int i = threadIdx.x; i < NN / 4; i += 256) {
    v4f v = __builtin_nontemporal_load(&m4[i]);
    acc += v.x + v.y + v.z + v.w;
  }
  __shared__ float red[256];
  red[threadIdx.x] = acc;
  __syncthreads();
  for (int off = 128; off > 0; off >>= 1) {
    if (threadIdx.x < off) red[threadIdx.x] += red[threadIdx.x + off];
    __syncthreads();
  }
  if (threadIdx.x == 0) deg[row] = red[0];
}

// coefT[l][j] = para_j * lab[j][l], para_j = row_j / sqrt(deg_j * deg_idx);
// also accumulates s_l = sum_j coefT[l][j].  l-major layout so the WMMA
// A-fragment (A[m][k], A[m][k+1]) is one contiguous b64 load.
__global__ __launch_bounds__(256) void coef_kernel(const float* __restrict__ matrix,
                                                   const int* __restrict__ labels,
                                                   const float* __restrict__ deg,
                                                   const int* __restrict__ index_p,
                                                   float* __restrict__ coefT,
                                                   float* __restrict__ s_out) {
  __shared__ float sblk[8];
  if (threadIdx.x < 8) sblk[threadIdx.x] = 0.f;
  __syncthreads();
  const int idx = index_p[0];
  const float dI = deg[idx];
  const int j = blockIdx.x * 256 + threadIdx.x;

  const float rj = matrix[(size_t)idx * NN + j];
  float para = 0.f;
  if (rj != 0.f) para = rj * rsqrtf(deg[j] * dI);

  float c[7];
#pragma unroll
  for (int l = 0; l < 7; ++l) c[l] = 0.f;
  if (para != 0.f) {
#pragma unroll
    for (int l = 0; l < 7; ++l) c[l] = para * (float)labels[j * 7 + l];
  }

#pragma unroll
  for (int l = 0; l < 7; ++l) coefT[(size_t)l * NN + j] = c[l];  // coalesced per l
  coefT[(size_t)7 * NN + j] = 0.f;

  if (para != 0.f) {
#pragma unroll
    for (int l = 0; l < 7; ++l)
      if (c[l] != 0.f) atomicAdd(&sblk[l], c[l]);
  }
  __syncthreads();
  if (threadIdx.x < 8) atomicAdd(&s_out[threadIdx.x], sblk[threadIdx.x]);
}

// partial[kc] += coefT (16 x 2048 slice) @ xf (2048 x 512 slice), f32 WMMA
// 16x16x4.  256 single-wave blocks: 32 N-tiles x 8 K-chunks.
// A frag (16x4 f32): lane = m + 16h; VGPR v holds A[m][k0 + 2h + v]  -> b64 load.
// B frag (4x16 f32): lane = n + 16h; VGPR v holds B[k0 + 2h + v][n].
// C/D (16x16 f32):   VGPR v, lane half h, lane n -> D[v + 8h][n].
__global__ __launch_bounds__(32) void agg_gemm_kernel(const float* __restrict__ coefT,
                                                      const float* __restrict__ xf,
                                                      float* __restrict__ partial) {
  const int nt = blockIdx.x & 31;       // N tile
  const int kc = blockIdx.x >> 5;       // K chunk
  const int nb = nt * 16;
  const int kbase = kc * KLEN;
  const int lane = threadIdx.x;
  const int m = lane & 15;              // == n for B/C addressing
  const int h = lane >> 4;

  v8f acc = {0.f, 0.f, 0.f, 0.f, 0.f, 0.f, 0.f, 0.f};
  const float* aP = coefT + (size_t)m * NN + kbase + 2 * h;    // A[m][k+2h..+1]
  const float* bP = xf + (size_t)(kbase + 2 * h) * DD + nb + m;
#pragma unroll 4
  for (int k0 = 0; k0 < KLEN; k0 += 4) {
    v2f a = *reinterpret_cast<const v2f*>(aP + k0);            // one b64 load
    v2f b;
    b.x = bP[(size_t)k0 * DD];
    b.y = bP[(size_t)k0 * DD + DD];
    acc = __builtin_amdgcn_wmma_f32_16x16x4_f32(false, a, false, b, (short)0, acc,
                                                false, false);
  }
#pragma unroll
  for (int v = 0; v < 8; ++v)
    partial[(size_t)kc * (16 * DD) + (v + 8 * h) * DD + nb + m] = acc[v];
}

// Deterministic split-K reduction: Amat[i] = sum_kc partial[kc][i].
__global__ __launch_bounds__(256) void reduce_kernel(const float* __restrict__ partial,
                                                     float* __restrict__ Amat) {
  const int i = blockIdx.x * 256 + threadIdx.x;   // 0 .. 16*512-1
  float acc = 0.f;
#pragma unroll
  for (int kc = 0; kc < KCH; ++kc) acc += partial[(size_t)kc * (16 * DD) + i];
  Amat[i] = acc;
}

// tmp_a = Amat[0:7,:] @ W2 + s (x) b2 ; h = relu(concat(z[idx], tmp_a)) ;
// out = max(x[idx], h) @ Wp + bp.   4 waves cover H=64 in 16-wide tiles.
__global__ __launch_bounds__(128) void final_kernel(const float* __restrict__ xf,
                                                    const float* __restrict__ W2,
                                                    const float* __restrict__ b2,
                                                    const float* __restrict__ Wp,
                                                    const float* __restrict__ bp,
                                                    const float* __restrict__ Amat,
                                                    const float* __restrict__ s,
                                                    const int* __restrict__ index_p,
                                                    float* __restrict__ out) {
  __shared__ float hbuf[512];
  const int tid = threadIdx.x;
  const int wave = tid >> 5;
  const int lane = tid & 31;
  const int m = lane & 15;
  const int h = lane >> 4;
  const int nb = wave * 16;             // output-H tile base

  v8f acc = {0.f, 0.f, 0.f, 0.f, 0.f, 0.f, 0.f, 0.f};
  const float* aP = Amat + m * DD + 2 * h;
  for (int k0 = 0; k0 < DD; k0 += 4) {
    v2f a = *reinterpret_cast<const v2f*>(aP + k0);            // one b64 load
    v2f b;
    b.x = W2[(k0 + 2 * h) * HH + nb + m];
    b.y = W2[(k0 + 2 * h + 1) * HH + nb + m];
    acc = __builtin_amdgcn_wmma_f32_16x16x4_f32(false, a, false, b, (short)0, acc,
                                                false, false);
  }
#pragma unroll
  for (int v = 0; v < 8; ++v) {
    const int l = v + 8 * h;
    if (l < 7) {
      const float t = acc[v] + s[l] * b2[nb + m];
      hbuf[64 + l * 64 + nb + m] = fmaxf(t, 0.f);   // relu(tmp_a) into concat slot
    }
  }

  const int idx = index_p[0];
  if (tid < 64) {                       // z[index] = xf[index] @ W2 + b2, relu
    float z = b2[tid];
    for (int k = 0; k < DD; ++k)
      z = fmaf(xf[(size_t)idx * DD + k], W2[k * HH + tid], z);
    hbuf[tid] = fmaxf(z, 0.f);
  }
  __syncthreads();

  if (tid < 7) {                        // out = max(xf[idx], h) @ Wp + bp
    float o = bp[tid];
    for (int d = 0; d < DD; ++d) {
      const float hc = fmaxf(xf[(size_t)idx * DD + d], hbuf[d]);
      o = fmaf(hc, Wp[d * 7 + tid], o);
    }
    out[tid] = o;
  }
}

extern "C" void kernel_launch(void* const* d_in, const int* in_sizes, int n_in,
                              void* d_out, int out_size, void* d_ws, size_t ws_size,
                              hipStream_t stream) {
  const float* matrix  = (const float*)d_in[0];
  const float* xf      = (const float*)d_in[1];
  const float* W2      = (const float*)d_in[2];
  const float* b2      = (const float*)d_in[3];
  const float* Wp      = (const float*)d_in[4];
  const float* bp      = (const float*)d_in[5];
  const int*   labels  = (const int*)d_in[6];
  const int*   index_p = (const int*)d_in[7];
  float* out = (float*)d_out;

  float* ws      = (float*)d_ws;
  float* deg     = ws;                   // 16384
  float* s       = ws + 16384;           // 16
  float* Amat    = ws + 16400;           // 16*512
  float* partial = ws + 24592;           // 8*16*512
  float* coefT   = ws + 90128;           // 16*16384 (rows 0..7 written)

  zero_s_kernel<<<1, 32, 0, stream>>>(s);
  deg_kernel<<<NN, 256, 0, stream>>>(matrix, deg);
  coef_kernel<<<NN / 256, 256, 0, stream>>>(matrix, labels, deg, index_p, coefT, s);
  agg_gemm_kernel<<<32 * KCH, 32, 0, stream>>>(coefT, xf, partial);
  reduce_kernel<<<(16 * DD) / 256, 256, 0, stream>>>(partial, Amat);
  final_kernel<<<1, 128, 0, stream>>>(xf, W2, b2, Wp, bp, Amat, s, index_p, out);
}